// PointConvBlock_64785286693674
// MI455X (gfx1250) — compile-verified
//
#include <hip/hip_runtime.h>
#include <hip/hip_bf16.h>

#define NPTS      16384
#define CIN       128
#define COUT      256
#define KNN_K     20
#define NEG_SLOPE 0.2f
#define BN_EPS    1e-5f

typedef float v2f __attribute__((ext_vector_type(2)));
typedef float v8f __attribute__((ext_vector_type(8)));
typedef int   b128v __attribute__((vector_size(16)));   // 16-byte transfer unit
typedef unsigned int v4u __attribute__((ext_vector_type(4)));
typedef int  v4i __attribute__((ext_vector_type(4)));
typedef int  v8i __attribute__((ext_vector_type(8)));

// ---- CDNA5 async global->LDS copy (ASYNCcnt path), with safe fallbacks ----
#if defined(__has_builtin) && __has_builtin(__builtin_amdgcn_global_load_async_to_lds_b128)
#define ASYNC_COPY_B128(gsrc, ldst)                                         \
    __builtin_amdgcn_global_load_async_to_lds_b128(                         \
        (__attribute__((address_space(1))) b128v*)(gsrc),                   \
        (__attribute__((address_space(3))) b128v*)(ldst), 0, 0)
#else
#define ASYNC_COPY_B128(gsrc, ldst) (*(float4*)(ldst) = *(const float4*)(gsrc))
#endif

#if defined(__has_builtin) && __has_builtin(__builtin_amdgcn_s_wait_asynccnt)
#define ASYNC_WAIT() __builtin_amdgcn_s_wait_asynccnt(0)
#else
#define ASYNC_WAIT() asm volatile("s_wait_asynccnt 0x0" ::: "memory")
#endif

// ---- CDNA5 Tensor Data Mover (TENSORcnt path) ----
#if defined(__has_builtin) && __has_builtin(__builtin_amdgcn_tensor_load_to_lds) && \
    defined(__has_builtin) && __has_builtin(__builtin_amdgcn_s_wait_tensorcnt)
#define PCB_HAVE_TDM 1
#else
#define PCB_HAVE_TDM 0
#endif

// ---------------------------------------------------------------------------
// Pass 0: squared norms of pos
// ---------------------------------------------------------------------------
__global__ __launch_bounds__(256) void pcb_sqnorm(const float* __restrict__ pos,
                                                  float* __restrict__ sq) {
    int i = blockIdx.x * 256 + threadIdx.x;
    if (i < NPTS) {
        float x = pos[3 * i + 0], y = pos[3 * i + 1], z = pos[3 * i + 2];
        sq[i] = x * x + y * y + z * z;
    }
}

// ---------------------------------------------------------------------------
// Pass 1: zero the per-channel stat accumulators
// ---------------------------------------------------------------------------
__global__ __launch_bounds__(256) void pcb_zero_stats(float* __restrict__ sums,
                                                      float* __restrict__ sumsq) {
    sums[threadIdx.x]  = 0.0f;
    sumsq[threadIdx.x] = 0.0f;
}

// ---------------------------------------------------------------------------
// Pass 2: brute-force KNN (top-20 smallest squared distances, self excluded).
// One thread per query; candidate tiles (256 x pos3 + 256 x sq) staged into
// LDS with async global->LDS b128 copies (ASYNCcnt). The 20-entry sorted
// list uses only compile-time indices -> stays in VGPRs.
// ---------------------------------------------------------------------------
__global__ __launch_bounds__(256) void pcb_knn(const float* __restrict__ pos,
                                               const float* __restrict__ sq,
                                               int* __restrict__ idx) {
    __shared__ float cpos[256 * 3];   // 3 KB
    __shared__ float csq[256];        // 1 KB
    const int tid = threadIdx.x;
    const int q = blockIdx.x * 256 + tid;
    const float qx = pos[3 * q + 0], qy = pos[3 * q + 1], qz = pos[3 * q + 2];
    const float qs = sq[q];

    float dl[KNN_K];
    int   il[KNN_K];
#pragma unroll
    for (int t = 0; t < KNN_K; ++t) { dl[t] = __builtin_inff(); il[t] = 0; }

    for (int base = 0; base < NPTS; base += 256) {
        // async-stage this candidate tile: 192 lanes move pos (3KB), 64 move sq (1KB)
        if (tid < 192) ASYNC_COPY_B128(&pos[(size_t)base * 3 + tid * 4], &cpos[tid * 4]);
        else           ASYNC_COPY_B128(&sq[base + (tid - 192) * 4], &csq[(tid - 192) * 4]);
        if (base + 256 < NPTS)  // hint next tile toward L2 (global_prefetch_b8)
            __builtin_prefetch(&pos[3 * (base + 256 + tid)], 0, 0);
        ASYNC_WAIT();
        __syncthreads();

        for (int t = 0; t < 256; ++t) {
            const float cx = cpos[3 * t + 0], cy = cpos[3 * t + 1], cz = cpos[3 * t + 2];
            const float d  = qs + csq[t] - 2.0f * (qx * cx + qy * cy + qz * cz);
            const int   j2 = base + t;
            if (j2 != q && d < dl[KNN_K - 1]) {
                // sorted-ascending insertion; condition chain is monotone so no
                // break needed and all indices are compile-time constants.
#pragma unroll
                for (int s = KNN_K - 1; s > 0; --s) {
                    if (dl[s - 1] > d)    { dl[s] = dl[s - 1]; il[s] = il[s - 1]; }
                    else if (dl[s] > d)   { dl[s] = d;         il[s] = j2; }
                }
                if (dl[0] > d) { dl[0] = d; il[0] = j2; }
            }
        }
        __syncthreads();
    }
#pragma unroll
    for (int t = 0; t < KNN_K; ++t) idx[q * KNN_K + t] = il[t];
}

// ---------------------------------------------------------------------------
// Pass 3: feature transform with f32 WMMA.
//   half==0 : abuf  = x @ W[0:128,:]   + bias
//   half==1 : bpbuf = x @ W[128:256,:]
// 256-thread block = 8 waves; block computes a 128x64 output tile. The shared
// 128x64 W panel is staged into LDS ONCE per block — preferably by a single
// Tensor Data Mover descriptor (TENSORcnt) issued by wave 0, else by per-lane
// async global->LDS copies. LDS row stride padded to 80 floats (TDM pad:
// interval 64 DW, amount 16 DW) so the K and K+2 rows (lanes 0-15 vs 16-31)
// hit disjoint banks.
// ---------------------------------------------------------------------------
#define WSTRIDE 80  // floats per LDS row (64 data + 16 pad)

__global__ __launch_bounds__(256) void pcb_gemm(const float* __restrict__ x,
                                                const float* __restrict__ W,
                                                const float* __restrict__ bias,
                                                float* __restrict__ abuf,
                                                float* __restrict__ bpbuf) {
    __shared__ float wlds[CIN * WSTRIDE];  // 40 KB

    const int tid   = threadIdx.x;
    const int lane  = tid & 31;
    const int wave  = tid >> 5;
    const int l15   = lane & 15;
    const int khalf = lane >> 4;           // 0: K pair {0,1}, 1: K pair {2,3}
    const int c0    = blockIdx.y * 64;
    const int half  = blockIdx.z;

    const float* Wsub = W + (size_t)half * CIN * COUT;
    float*       outp = half ? bpbuf : abuf;

    // ---- stage W[0:128, c0:c0+64] into LDS ----
#if PCB_HAVE_TDM
    if (wave == 0) {
        const unsigned long long ga = (unsigned long long)(const void*)(Wsub + c0);
        const unsigned lds_base = (unsigned)(unsigned long long)(const void*)&wlds[0];
        v4u g0;
        g0[0] = 1u;                                    // count=1 valid user D#
        g0[1] = lds_base;                              // lds_addr
        g0[2] = (unsigned)(ga & 0xFFFFFFFFu);          // global_addr[31:0]
        g0[3] = (unsigned)((ga >> 32) & 0x01FFFFFFu)   // global_addr[56:32]
              | 0x80000000u;                           // type=2 (image)
        v8i g1;
        g1[0] = (int)((2u << 16)      // data_size = 4B
                    | (1u << 20)      // pad_enable
                    | (5u << 22)      // pad_interval: 64 DWORDs
                    | (15u << 25));   // pad_amount: 16 DWORDs
        g1[1] = (int)(256u << 16);    // tensor_dim0 = 256 (low 16)
        g1[2] = (int)(128u << 16);    // tensor_dim0 hi | tensor_dim1 = 128 (low 16)
        g1[3] = (int)(64u << 16);     // tensor_dim1 hi | tile_dim0 = 64
        g1[4] = 128;                  // tile_dim1 = 128, tile_dim2 = 0
        g1[5] = 256;                  // tensor_dim0_stride[31:0] = 256
        g1[6] = 0;                    // stride hi | tensor_dim1_stride lo
        g1[7] = 0;
        v4i gz  = {0, 0, 0, 0};
        v8i gz8 = {0, 0, 0, 0, 0, 0, 0, 0};
        __builtin_amdgcn_tensor_load_to_lds(g0, g1, gz, gz, gz8, 0);
        __builtin_amdgcn_s_wait_tensorcnt(0);
    }
    __syncthreads();
#else
#pragma unroll
    for (int it = 0; it < 8; ++it) {
        const int e4 = it * 256 + tid;     // float4 index, 0..2047
        const int k  = e4 >> 4;            // K row (16 float4 per row)
        const int j4 = e4 & 15;            // float4 within row
        ASYNC_COPY_B128(Wsub + (size_t)k * COUT + c0 + j4 * 4,
                        &wlds[k * WSTRIDE + j4 * 4]);
    }
    ASYNC_WAIT();
    __syncthreads();
#endif

    // ---- per-wave 16x64 WMMA tile ----
    const int m0 = blockIdx.x * 128 + wave * 16;
    const float* xrow = x + (size_t)(m0 + l15) * CIN;

    v8f acc0 = {}, acc1 = {}, acc2 = {}, acc3 = {};
    for (int k0 = 0; k0 < CIN; k0 += 4) {
        const int kk = k0 + 2 * khalf;
        v2f A;
        A.x = xrow[kk];
        A.y = xrow[kk + 1];
        const float* w0 = &wlds[kk * WSTRIDE + l15];
        const float* w1 = w0 + WSTRIDE;
        v2f B0, B1, B2, B3;
        B0.x = w0[0];  B0.y = w1[0];
        B1.x = w0[16]; B1.y = w1[16];
        B2.x = w0[32]; B2.y = w1[32];
        B3.x = w0[48]; B3.y = w1[48];
        acc0 = __builtin_amdgcn_wmma_f32_16x16x4_f32(false, A, false, B0, (short)0, acc0, false, false);
        acc1 = __builtin_amdgcn_wmma_f32_16x16x4_f32(false, A, false, B1, (short)0, acc1, false, false);
        acc2 = __builtin_amdgcn_wmma_f32_16x16x4_f32(false, A, false, B2, (short)0, acc2, false, false);
        acc3 = __builtin_amdgcn_wmma_f32_16x16x4_f32(false, A, false, B3, (short)0, acc3, false, false);
    }

    float b0 = 0.f, b1 = 0.f, b2 = 0.f, b3 = 0.f;
    if (half == 0) {
        b0 = bias[c0 + l15];
        b1 = bias[c0 + 16 + l15];
        b2 = bias[c0 + 32 + l15];
        b3 = bias[c0 + 48 + l15];
    }
    // C/D layout: VGPR v -> row m0 + v + 8*khalf, lane l15 -> column
#pragma unroll
    for (int v = 0; v < 8; ++v) {
        float* orp = outp + (size_t)(m0 + v + 8 * khalf) * COUT + c0 + l15;
        orp[0]  = acc0[v] + b0;
        orp[16] = acc1[v] + b1;
        orp[32] = acc2[v] + b2;
        orp[48] = acc3[v] + b3;
    }
}

// ---------------------------------------------------------------------------
// Pass 4: single edge pass. thread = channel, block handles 16 queries.
// Produces per-(n,c) running max AND min of raw h, plus per-channel sum/sumsq
// (block-local registers -> one atomic pair per thread).  idx loads are
// threadIdx-independent -> scalarized (SMEM); bp gathers are L2-resident.
// ---------------------------------------------------------------------------
#define QPB 16
__global__ __launch_bounds__(256) void pcb_edges(const float* __restrict__ abuf,
                                                 const float* __restrict__ bpbuf,
                                                 const int* __restrict__ idx,
                                                 float* __restrict__ hmax,
                                                 float* __restrict__ hmin,
                                                 float* __restrict__ sums,
                                                 float* __restrict__ sumsq) {
    const int c  = threadIdx.x;
    const int q0 = blockIdx.x * QPB;
    float lsum = 0.f, lsq = 0.f;
    for (int qi = 0; qi < QPB; ++qi) {
        const int   q  = q0 + qi;
        const float av = abuf[(size_t)q * COUT + c];
        float mx = -__builtin_inff(), mn = __builtin_inff();
#pragma unroll
        for (int k = 0; k < KNN_K; ++k) {
            const int   j = idx[q * KNN_K + k];
            const float h = av + bpbuf[(size_t)j * COUT + c];
            mx   = fmaxf(mx, h);
            mn   = fminf(mn, h);
            lsum += h;
            lsq   = fmaf(h, h, lsq);
        }
        hmax[(size_t)q * COUT + c] = mx;
        hmin[(size_t)q * COUT + c] = mn;
    }
    atomicAdd(&sums[c], lsum);
    atomicAdd(&sumsq[c], lsq);
}

// ---------------------------------------------------------------------------
// Pass 5: per-channel scale/shift from batch statistics
// ---------------------------------------------------------------------------
__global__ __launch_bounds__(256) void pcb_finalize_stats(const float* __restrict__ sums,
                                                          const float* __restrict__ sumsq,
                                                          const float* __restrict__ gamma,
                                                          const float* __restrict__ beta,
                                                          float* __restrict__ scale,
                                                          float* __restrict__ shift) {
    const int   c    = threadIdx.x;
    const float inv  = 1.0f / (float)(NPTS * KNN_K);
    const float mean = sums[c] * inv;
    const float var  = sumsq[c] * inv - mean * mean;
    const float s    = gamma[c] * rsqrtf(var + BN_EPS);
    scale[c] = s;
    shift[c] = beta[c] - mean * s;
}

// ---------------------------------------------------------------------------
// Pass 6: normalization + LeakyReLU + neighbor max.  scale>=0 -> use hmax,
// scale<0 -> use hmin (affine map + monotone activation commute with max).
// ---------------------------------------------------------------------------
__global__ __launch_bounds__(256) void pcb_output(const float* __restrict__ hmax,
                                                  const float* __restrict__ hmin,
                                                  const float* __restrict__ scale,
                                                  const float* __restrict__ shift,
                                                  float* __restrict__ out) {
    const int   i = blockIdx.x * 256 + threadIdx.x;   // over NPTS*COUT
    const int   c = i & (COUT - 1);
    const float s = scale[c];
    const float h = (s >= 0.0f) ? hmax[i] : hmin[i];
    const float v = fmaf(s, h, shift[c]);
    out[i] = (v >= 0.0f) ? v : NEG_SLOPE * v;
}

// ---------------------------------------------------------------------------
extern "C" void kernel_launch(void* const* d_in, const int* in_sizes, int n_in,
                              void* d_out, int out_size, void* d_ws, size_t ws_size,
                              hipStream_t stream) {
    (void)in_sizes; (void)n_in; (void)out_size; (void)ws_size;

    const float* pos   = (const float*)d_in[0];
    const float* x     = (const float*)d_in[1];
    const float* W     = (const float*)d_in[2];
    const float* bias  = (const float*)d_in[3];
    const float* gamma = (const float*)d_in[4];
    const float* beta  = (const float*)d_in[5];
    float*       out   = (float*)d_out;

    char* ws = (char*)d_ws;
    float* sq    = (float*)(ws);                              // 64 KB
    float* scale = (float*)(ws + (1 << 16));
    float* shift = (float*)(ws + (1 << 16) + 1024);
    float* sums  = (float*)(ws + (1 << 16) + 2048);
    float* sumsq = (float*)(ws + (1 << 16) + 3072);
    int*   idx   = (int*)  (ws + ((size_t)1 << 20));          // 1.25 MB
    float* abuf  = (float*)(ws + ((size_t)4  << 20));         // 16 MB
    float* bpbuf = (float*)(ws + ((size_t)20 << 20));         // 16 MB
    float* hmax  = (float*)(ws + ((size_t)36 << 20));         // 16 MB
    float* hmin  = (float*)(ws + ((size_t)52 << 20));         // 16 MB

    pcb_sqnorm<<<NPTS / 256, 256, 0, stream>>>(pos, sq);
    pcb_zero_stats<<<1, 256, 0, stream>>>(sums, sumsq);
    pcb_knn<<<NPTS / 256, 256, 0, stream>>>(pos, sq, idx);
    pcb_gemm<<<dim3(NPTS / 128, COUT / 64, 2), 256, 0, stream>>>(x, W, bias, abuf, bpbuf);
    pcb_edges<<<NPTS / QPB, 256, 0, stream>>>(abuf, bpbuf, idx, hmax, hmin, sums, sumsq);
    pcb_finalize_stats<<<1, 256, 0, stream>>>(sums, sumsq, gamma, beta, scale, shift);
    pcb_output<<<(NPTS * COUT) / 256, 256, 0, stream>>>(hmax, hmin, scale, shift, out);
}